// AssociativeBinding_42245298323623
// MI455X (gfx1250) — compile-verified
//
#include <hip/hip_runtime.h>
#include <math.h>

typedef __attribute__((ext_vector_type(2))) float v2f;
typedef __attribute__((ext_vector_type(4))) float v4f;
typedef __attribute__((ext_vector_type(8))) float v8f;

#define MDIM 64
#define MEM_PER_B (MDIM * MDIM * MDIM) /* 262144 floats = 1 MB */
#define HDIM 512
#define NTHREADS 512                   /* 16 waves: 4 f-tiles x 4 rt-quarters */

__global__ __launch_bounds__(NTHREADS) void assoc_bind_wmma_kernel(
    const float* __restrict__ mem,    // [B,64,64,64]
    const float* __restrict__ hid,    // [B,512]
    const float* __restrict__ role1,  // [B,64]
    const float* __restrict__ role2,  // [B,64]
    const float* __restrict__ filer,  // [B,64]
    const float* __restrict__ Wg,     // [1,512]
    const float* __restrict__ bg,     // [1]
    float* __restrict__ out)          // [B,64,64,64]
{
    const int b   = blockIdx.x;
    const int tid = threadIdx.x;

    __shared__ __align__(16) float s_role1[64];
    __shared__ __align__(16) float s_role2[64];
    __shared__ __align__(16) float s_prev[64];
    __shared__ __align__(16) float s_cfin[64];
    // [0]=h·W, [1]=sum(mem^2), [2]=<ci,prev>, [3]=|ci|^2, [4]=|r1|^2, [5]=|r2|^2, [6]=1/norm
    __shared__ float s_red[8];

    const float* memb = mem + (size_t)b * MEM_PER_B;
    float*       outb = out + (size_t)b * MEM_PER_B;

    if (tid < 64) {
        s_role1[tid] = role1[b * 64 + tid];
        s_role2[tid] = role2[b * 64 + tid];
        s_prev[tid]  = 0.0f;
    }
    if (tid < 8) s_red[tid] = 0.0f;
    __syncthreads();

    // ---- tiny reductions: write-gate dot product, |role1|^2, |role2|^2 ----
    {
        float hp = hid[b * HDIM + tid] * Wg[tid];   // 512 threads x 1 element
        atomicAdd(&s_red[0], hp);
        if (tid < 64) {
            atomicAdd(&s_red[4], s_role1[tid] * s_role1[tid]);
            atomicAdd(&s_red[5], s_role2[tid] * s_role2[tid]);
        }
    }

    // ---- pass 1: prev[f] = sum_rt w[rt]*mem[rt,f] via V_WMMA_F32_16X16X4_F32,
    //      plus sum(mem^2) off the same loads ------------------------------------
    const int wave     = tid >> 5;
    const int lane     = tid & 31;
    const int f_base   = (wave & 3) * 16;        // which 16 f-columns this wave owns
    const int rt_begin = (wave >> 2) * 1024;     // which rt-quarter this wave owns
    const int col      = f_base + (lane & 15);   // A layout: lane -> M (= f index)
    const int kk       = (lane >> 4) * 2;        // A layout: half-wave -> K offset {0,2}

    v8f   acc0 = {};
    v8f   acc1 = {};
    float ssq  = 0.0f;

    for (int c = 0; c < 256; c += 2) {
        {
            const int rt0 = rt_begin + c * 4;
            const float* p = memb + (size_t)(rt0 + kk) * 64 + col;
            const float a0 = p[0];
            const float a1 = p[64];
            ssq += a0 * a0 + a1 * a1;
            const float  r1v = s_role1[rt0 >> 6];
            const float2 r2  = *reinterpret_cast<const float2*>(&s_role2[(rt0 & 63) + kk]);
            v2f A = {a0, a1};
            v2f B = {r1v * r2.x, r1v * r2.y};   // w broadcast into all 16 columns
            acc0 = __builtin_amdgcn_wmma_f32_16x16x4_f32(
                false, A, false, B, (short)0, acc0, false, false);
        }
        {
            const int rt0 = rt_begin + (c + 1) * 4;
            const float* p = memb + (size_t)(rt0 + kk) * 64 + col;
            const float a0 = p[0];
            const float a1 = p[64];
            ssq += a0 * a0 + a1 * a1;
            const float  r1v = s_role1[rt0 >> 6];
            const float2 r2  = *reinterpret_cast<const float2*>(&s_role2[(rt0 & 63) + kk]);
            v2f A = {a0, a1};
            v2f B = {r1v * r2.x, r1v * r2.y};
            acc1 = __builtin_amdgcn_wmma_f32_16x16x4_f32(
                false, A, false, B, (short)0, acc1, false, false);
        }
    }

    atomicAdd(&s_red[1], ssq);

    // All 16 D columns are identical (B columns were broadcast); column-0 lanes
    // (lane 0 -> rows M=0..7, lane 16 -> rows M=8..15) scatter into LDS prev[].
    {
        v8f acc = acc0 + acc1;
        if ((lane & 15) == 0) {
            union { v8f v; float f[8]; } u;
            u.v = acc;
            const int mbase = f_base + (lane >> 4) * 8;
#pragma unroll
            for (int j = 0; j < 8; ++j)
                atomicAdd(&s_prev[mbase + j], u.f[j]);
        }
    }
    __syncthreads();

    // ---- pass 2 prep: gate, cur_info, analytic norm --------------------------
    if (tid < 64) {
        const float gate = 1.0f / (1.0f + expf(-(s_red[0] + bg[0] + 1.0f)));
        const float ci   = gate * (filer[b * 64 + tid] - s_prev[tid]);
        s_cfin[tid] = ci;
        atomicAdd(&s_red[2], ci * s_prev[tid]);
        atomicAdd(&s_red[3], ci * ci);
    }
    __syncthreads();
    if (tid == 0) {
        const float cs  = 1.0f / 64.0f;
        // ||new||^2 = ||mem||^2 + 2*cs*<ci,prev> + cs^2*||r1||^2*||r2||^2*||ci||^2
        float nsq = s_red[1] + 2.0f * cs * s_red[2]
                  + cs * cs * s_red[4] * s_red[5] * s_red[3];
        float nrm = sqrtf(fmaxf(nsq, 0.0f));
        nrm       = fmaxf(nrm, 1.0f);   // relu(n-1)+1
        s_red[6]  = 1.0f / nrm;
    }
    __syncthreads();
    const float inv = s_red[6];
    if (tid < 64) s_cfin[tid] *= (1.0f / 64.0f) * inv;
    __syncthreads();

    // ---- pass 2: out = mem*inv + (role1[r]*role2[t]) * cfin[f], b128 streams.
    //      mem read is last-use -> non-temporal; out store is write-once -> NT,
    //      preserving L2 for the pass1->pass2 tile reuse of other blocks. -------
    const int   f4  = (tid * 4) & 63;
    const float cf0 = s_cfin[f4 + 0];
    const float cf1 = s_cfin[f4 + 1];
    const float cf2 = s_cfin[f4 + 2];
    const float cf3 = s_cfin[f4 + 3];

    for (int it = 0; it < MEM_PER_B / (NTHREADS * 4); ++it) {
        const int idx = (it * NTHREADS + tid) * 4;     // element index, f4-aligned
        const int rt  = idx >> 6;
        const float w = s_role1[rt >> 6] * s_role2[rt & 63];
        const v4f m4 =
            __builtin_nontemporal_load(reinterpret_cast<const v4f*>(memb + idx));
        v4f o;
        o.x = fmaf(m4.x, inv, w * cf0);
        o.y = fmaf(m4.y, inv, w * cf1);
        o.z = fmaf(m4.z, inv, w * cf2);
        o.w = fmaf(m4.w, inv, w * cf3);
        __builtin_nontemporal_store(o, reinterpret_cast<v4f*>(outb + idx));
    }
}

extern "C" void kernel_launch(void* const* d_in, const int* in_sizes, int n_in,
                              void* d_out, int out_size, void* d_ws, size_t ws_size,
                              hipStream_t stream) {
    const float* mem   = (const float*)d_in[0];
    const float* hid   = (const float*)d_in[1];
    const float* role1 = (const float*)d_in[2];
    const float* role2 = (const float*)d_in[3];
    const float* filer = (const float*)d_in[4];
    const float* Wg    = (const float*)d_in[5];
    const float* bg    = (const float*)d_in[6];
    float*       out   = (float*)d_out;

    const int B = in_sizes[2] / 64;   // role1 is [B,64]
    assoc_bind_wmma_kernel<<<dim3(B), dim3(NTHREADS), 0, stream>>>(
        mem, hid, role1, role2, filer, Wg, bg, out);
}